// O3Linear_30691836297240
// MI455X (gfx1250) — compile-verified
//
#include <hip/hip_runtime.h>

// ---- CDNA5 (gfx1250) O(3)-equivariant block-diagonal linear, fp32 WMMA ----
// y[s,v,i] = sum_u x[s,u,i] * w[u,v] for blocks (mul,dim)=(64,1),(32,3),(16,5),(8,7)
// Memory-bound (AI ~7.5 FLOP/B): async global<->LDS streaming (ASYNCcnt path),
// math on v_wmma_f32_16x16x4_f32 fed by ds_load.

typedef float v2f __attribute__((ext_vector_type(2)));
typedef float v8f __attribute__((ext_vector_type(8)));

#define ROW        296                 // floats per sample row
#define ROW4       74                  // float4 per sample row
#define S_TILE     64                  // samples per workgroup
#define PITCH      300                 // LDS x/y tile row pitch (floats); 44*m mod 64 injective -> no bank conflicts
#define XTILE      (S_TILE * PITCH)    // 19200 floats
#define WOFF0      XTILE               // w0: 64x65 padded
#define WOFF1      (WOFF0 + 64 * 65)   // w1: 32x33
#define WOFF2      (WOFF1 + 32 * 33)   // w2: 16x17
#define WOFF3      (WOFF2 + 16 * 17)   // w3: 8x9
#define LDS_FLOATS (WOFF3 + 8 * 9)     // 24760 floats = 99,040 B -> 2 WG / WGP
#define THREADS    512
#define WAVES      (THREADS / 32)      // wave32: 16 waves
#define NTILES     88                  // 16 + 24 + 20 + 28 output tiles per 64-sample chunk
#define MAXQ       ((NTILES + WAVES - 1) / WAVES) // 6 tiles max per wave

// ---- per-tile WMMA accumulate: D += A(xtile rows) * B(weights) ----
// A f32 16x4: V0 = K=(0|2) by half-wave, V1 = K=(1|3); M = lane&15.
// B f32 4x16: V0 = K=(0|2) by half-wave, V1 = K=(1|3); N = lane&15.
template <int MUL, int DIM, int OFF, int WOFF>
__device__ __forceinline__ void tile_mma(const float* lds, int rt, int vt,
                                         int m, int half, v8f& acc) {
  const int row = rt * 16 + m;          // A row for this lane (both K pairs)
  const int s   = row / DIM;
  const int i   = row % DIM;
  const float* xr = lds + s * PITCH + OFF + i;         // stride DIM per k
  const int nabs = vt * 16 + m;                        // B column (n = lane&15 == m)
  const bool nvalid = (MUL >= 16) ? true : (nabs < MUL);
  const float* wc = lds + WOFF + nabs;                 // stride (MUL+1) per k
#pragma unroll
  for (int k0 = 0; k0 < MUL; k0 += 4) {
    const int ka = k0 + 2 * half;
    v2f a, b;
    a.x = xr[(ka + 0) * DIM];
    a.y = xr[(ka + 1) * DIM];
    b.x = nvalid ? wc[(ka + 0) * (MUL + 1)] : 0.0f;
    b.y = nvalid ? wc[(ka + 1) * (MUL + 1)] : 0.0f;
    acc = __builtin_amdgcn_wmma_f32_16x16x4_f32(false, a, false, b,
                                                (short)0, acc, false, false);
  }
}

// ---- write D fragment into the LDS tile (x-layout reused as y-layout) ----
// D 16x16 f32: VGPR j -> M = j + 8*half, N = lane&15.
template <int MUL, int DIM, int OFF>
__device__ __forceinline__ void tile_store(float* lds, int rt, int vt,
                                           int m, int half, const v8f& acc) {
  const int nabs = vt * 16 + m;
  if (MUL < 16 && nabs >= MUL) return;  // block 3: v padded 8 -> 16
#pragma unroll
  for (int j = 0; j < 8; ++j) {
    const int row = rt * 16 + j + 8 * half;
    const int s   = row / DIM;
    const int i   = row % DIM;
    lds[s * PITCH + OFF + nabs * DIM + i] = acc[j];
  }
}

__global__ __launch_bounds__(THREADS) void o3lin_wmma_kernel(
    const float* __restrict__ x, const float* __restrict__ w0,
    const float* __restrict__ w1, const float* __restrict__ w2,
    const float* __restrict__ w3, float* __restrict__ out, int nsamp) {
  extern __shared__ float lds[];
  const int tid  = threadIdx.x;
  const int lane = tid & 31;
  const int wave = tid >> 5;
  const long long sbase = (long long)blockIdx.x * S_TILE;
  // Dynamic LDS begins right after static LDS; our tile is at its base.
  const unsigned lds_base = __builtin_amdgcn_groupstaticsize();

  // ---- Phase A: stage weights (bank-padded) and x tile into LDS ----
  for (int idx = tid; idx < 64 * 64; idx += THREADS) {
    int u = idx >> 6, v = idx & 63;
    lds[WOFF0 + u * 65 + v] = w0[idx];
  }
  for (int idx = tid; idx < 32 * 32; idx += THREADS) {
    int u = idx >> 5, v = idx & 31;
    lds[WOFF1 + u * 33 + v] = w1[idx];
  }
  for (int idx = tid; idx < 16 * 16; idx += THREADS) {
    int u = idx >> 4, v = idx & 15;
    lds[WOFF2 + u * 17 + v] = w2[idx];
  }
  if (tid < 64) {
    int u = tid >> 3, v = tid & 7;
    lds[WOFF3 + u * 9 + v] = w3[tid];
  }

  // x tile: direct cache->LDS async b128 copies (ASYNCcnt), no VGPR round trip.
  const int tile_f4 = S_TILE * ROW4;  // 4736 float4s
  for (int idx = tid; idx < tile_f4; idx += THREADS) {
    const int s  = idx / ROW4;
    const int c4 = idx - s * ROW4;
    long long gs = sbase + s;
    if (gs >= nsamp) gs = nsamp - 1;  // clamp: keep EXEC full, LDS deterministic
    const unsigned lds_b = lds_base + 4u * (unsigned)(s * PITCH + c4 * 4);
    const unsigned long long ga =
        (unsigned long long)(const void*)(((const float4*)x) + gs * ROW4 + c4);
    asm volatile("global_load_async_to_lds_b128 %0, %1, off"
                 :: "v"(lds_b), "v"(ga) : "memory");
  }
  asm volatile("s_wait_asynccnt 0" ::: "memory");
  __syncthreads();

  // ---- Phase B: WMMA compute; wave w owns tiles w, w+16, ... (uniform) ----
  const int m    = lane & 15;
  const int half = lane >> 4;
  v8f acc[MAXQ];
  int tb[MAXQ], trt[MAXQ], tvt[MAXQ];
#pragma unroll
  for (int q = 0; q < MAXQ; ++q) {
    acc[q] = (v8f){};
    const int t = wave + q * WAVES;
    if (t < NTILES) {
      int b, rt, vt;
      if (t < 16)      { b = 0; rt = t >> 2;  vt = t & 3; }
      else if (t < 40) { int u = t - 16; b = 1; rt = u >> 1; vt = u & 1; }
      else if (t < 60) { b = 2; rt = t - 40; vt = 0; }
      else             { b = 3; rt = t - 60; vt = 0; }
      tb[q] = b; trt[q] = rt; tvt[q] = vt;
      switch (b) {
        case 0: tile_mma<64, 1,   0, WOFF0>(lds, rt, vt, m, half, acc[q]); break;
        case 1: tile_mma<32, 3,  64, WOFF1>(lds, rt, vt, m, half, acc[q]); break;
        case 2: tile_mma<16, 5, 160, WOFF2>(lds, rt, vt, m, half, acc[q]); break;
        default: tile_mma<8, 7, 240, WOFF3>(lds, rt, vt, m, half, acc[q]); break;
      }
    } else {
      tb[q] = -1; trt[q] = 0; tvt[q] = 0;
    }
  }
  __syncthreads();  // all reads of x-tile done before overwriting with y

  // ---- write D fragments into the (reused) LDS tile ----
#pragma unroll
  for (int q = 0; q < MAXQ; ++q) {
    if (tb[q] < 0) continue;
    switch (tb[q]) {
      case 0: tile_store<64, 1,   0>(lds, trt[q], tvt[q], m, half, acc[q]); break;
      case 1: tile_store<32, 3,  64>(lds, trt[q], tvt[q], m, half, acc[q]); break;
      case 2: tile_store<16, 5, 160>(lds, trt[q], tvt[q], m, half, acc[q]); break;
      default: tile_store<8, 7, 240>(lds, trt[q], tvt[q], m, half, acc[q]); break;
    }
  }
  __syncthreads();

  // ---- Phase C: async LDS->global b128 stores of the y tile ----
  for (int idx = tid; idx < tile_f4; idx += THREADS) {
    const int s  = idx / ROW4;
    const int c4 = idx - s * ROW4;
    const long long gs = sbase + s;
    if (gs < nsamp) {
      const unsigned lds_b = lds_base + 4u * (unsigned)(s * PITCH + c4 * 4);
      const unsigned long long ga =
          (unsigned long long)(void*)(((float4*)out) + gs * ROW4 + c4);
      asm volatile("global_store_async_from_lds_b128 %0, %1, off"
                   :: "v"(ga), "v"(lds_b) : "memory");
    }
  }
  asm volatile("s_wait_asynccnt 0" ::: "memory");
}

extern "C" void kernel_launch(void* const* d_in, const int* in_sizes, int n_in,
                              void* d_out, int out_size, void* d_ws,
                              size_t ws_size, hipStream_t stream) {
  const float* x  = (const float*)d_in[0];
  const float* w0 = (const float*)d_in[1];
  const float* w1 = (const float*)d_in[2];
  const float* w2 = (const float*)d_in[3];
  const float* w3 = (const float*)d_in[4];
  float* out = (float*)d_out;
  const int nsamp = in_sizes[0] / ROW;
  const int nblk  = (nsamp + S_TILE - 1) / S_TILE;
  const size_t shmem = (size_t)LDS_FLOATS * sizeof(float);
  o3lin_wmma_kernel<<<nblk, THREADS, shmem, stream>>>(x, w0, w1, w2, w3, out,
                                                      nsamp);
}